// NeuralODEWarp_3427383902477
// MI455X (gfx1250) — compile-verified
//
#include <hip/hip_runtime.h>
#include <cstdint>

#define TPB 256
#define NW 8               // waves per block (wave32)
#define EV_PER_BLOCK 128   // 16 events per wave
#define MAX_STEPS 64
#define HSTRIDE 68         // column stride (floats) for activation tiles:
                           // 68%4==0 -> 16B-aligned float4 ops, 68%64==4 -> bank spread

typedef float v2f __attribute__((ext_vector_type(2)));
typedef float v8f __attribute__((ext_vector_type(8)));

struct OdeState { float t; float dt; unsigned int done; unsigned int cur; };

// Dopri5 tableau (f32, computed in double then truncated)
__device__ __constant__ float d_C[7] = {0.0f, 0.2f, 0.3f, 0.8f, (float)(8.0/9.0), 1.0f, 1.0f};
__device__ __constant__ float d_A[7][6] = {
  {0,0,0,0,0,0},
  {(float)(1.0/5.0),0,0,0,0,0},
  {(float)(3.0/40.0),(float)(9.0/40.0),0,0,0,0},
  {(float)(44.0/45.0),(float)(-56.0/15.0),(float)(32.0/9.0),0,0,0},
  {(float)(19372.0/6561.0),(float)(-25360.0/2187.0),(float)(64448.0/6561.0),(float)(-212.0/729.0),0,0},
  {(float)(9017.0/3168.0),(float)(-355.0/33.0),(float)(46732.0/5247.0),(float)(49.0/176.0),(float)(-5103.0/18656.0),0},
  {(float)(35.0/384.0),0.0f,(float)(500.0/1113.0),(float)(125.0/192.0),(float)(-2187.0/6784.0),(float)(11.0/84.0)}
};
__device__ __constant__ float d_B5[7] = {
  (float)(35.0/384.0), 0.0f, (float)(500.0/1113.0), (float)(125.0/192.0),
  (float)(-2187.0/6784.0), (float)(11.0/84.0), 0.0f
};
__device__ __constant__ float d_E[7] = {
  (float)(35.0/384.0 - 5179.0/57600.0), 0.0f,
  (float)(500.0/1113.0 - 7571.0/16695.0),
  (float)(125.0/192.0 - 393.0/640.0),
  (float)(-2187.0/6784.0 + 92097.0/339200.0),
  (float)(11.0/84.0 - 187.0/2100.0),
  (float)(-1.0/40.0)
};

__device__ __forceinline__ v8f wmma_f32(v2f a, v2f b, v8f c) {
  // V_WMMA_F32_16X16X4_F32: D(16x16 f32) = A(16x4 f32) x B(4x16 f32) + C
  return __builtin_amdgcn_wmma_f32_16x16x4_f32(false, a, false, b, (short)0, c, false, false);
}

__device__ __forceinline__ float relu(float x) {
  // Single v_max_num_f32 (IEEE maximumNumber: quiets NaN itself) — avoids the
  // canonicalize max(x,x) the compiler pairs with llvm.maxnum on wmma results.
  float r;
  asm("v_max_num_f32 %0, 0, %1" : "=v"(r) : "v"(x));
  return r;
}

// One MLP evaluation for a wave's 16-event tile.
// A-fragments (all weights) live in registers; activations bounce through LDS
// column-major tiles (elem(row,col) at [col*stride + row]) so B-fragments are
// single b64 loads and C writebacks are b128 stores. Biases are loaded from LDS
// directly into the WMMA accumulator registers (b128 loads, no post-add).
// lane split: ln = lane&15 (event column / A-row), hi = lane>>4 (WMMA K/M half).
__device__ __forceinline__ void odefunc(
    float tc, const float yi[3], float kout[3],
    const v2f aw1[4], const v2f aw2[16][4], const v2f aw3[16],
    const float* sB1, const float* sB2, const float* sB3,
    float* sX, float* sH, int ln, int hi)
{
  // Build X tile (4 rows x 16 cols), column-major: one float4 store per event
  if (hi == 0) {
    *(float4*)&sX[ln*4] = make_float4(yi[0], yi[1], yi[2], tc);
  }
  v2f bx = *(const v2f*)&sX[ln*4 + 2*hi];   // rows (2hi, 2hi+1), col ln

  // ---- layer 1: H1 = relu(W1(64x4) @ X(4x16) + b1)
  #pragma unroll
  for (int m = 0; m < 4; ++m) {
    const float* bp = &sB1[m*16 + 8*hi];
    v8f c;
    #pragma unroll
    for (int r = 0; r < 8; ++r) c[r] = bp[r];      // contiguous 32B, b128-merged
    c = wmma_f32(aw1[m], bx, c);
    int base = ln*HSTRIDE + m*16 + 8*hi;           // 8 consecutive rows, col ln
    *(float4*)&sH[base+0] = make_float4(relu(c[0]), relu(c[1]), relu(c[2]), relu(c[3]));
    *(float4*)&sH[base+4] = make_float4(relu(c[4]), relu(c[5]), relu(c[6]), relu(c[7]));
  }

  // ---- layer 2: H2 = relu(W2(64x64) @ H1(64x16) + b2)
  // all 4 M-tiles accumulated in registers so sH can be overwritten afterwards
  v8f accs[4];
  #pragma unroll
  for (int m = 0; m < 4; ++m) {
    const float* bp = &sB2[m*16 + 8*hi];
    #pragma unroll
    for (int r = 0; r < 8; ++r) accs[m][r] = bp[r];
  }
  #pragma unroll
  for (int k = 0; k < 16; ++k) {
    v2f b = *(const v2f*)&sH[ln*HSTRIDE + 4*k + 2*hi];   // single b64
    #pragma unroll
    for (int m = 0; m < 4; ++m) {
      accs[m] = wmma_f32(aw2[k][m], b, accs[m]);         // A from registers
    }
  }
  #pragma unroll
  for (int m = 0; m < 4; ++m) {
    int base = ln*HSTRIDE + m*16 + 8*hi;
    v8f c = accs[m];
    *(float4*)&sH[base+0] = make_float4(relu(c[0]), relu(c[1]), relu(c[2]), relu(c[3]));
    *(float4*)&sH[base+4] = make_float4(relu(c[4]), relu(c[5]), relu(c[6]), relu(c[7]));
  }

  // ---- layer 3: out = W3pad(16x64, rows 3..15 zero) @ H2(64x16) + b3pad
  v8f c3;
  {
    const float* bp = &sB3[8*hi];                  // b3 in rows 0..2, 0 elsewhere
    #pragma unroll
    for (int r = 0; r < 8; ++r) c3[r] = bp[r];
  }
  #pragma unroll
  for (int k = 0; k < 16; ++k) {
    v2f b = *(const v2f*)&sH[ln*HSTRIDE + 4*k + 2*hi];
    c3 = wmma_f32(aw3[k], b, c3);
  }
  // hi==0 lanes: rows 0..2 (the real output). hi==1 lanes: rows 8..10 of the
  // zero-padded W3/b3 -> exactly 0. Those lanes' kout/yi are never consumed
  // (X-tile store, y5 store and error sum are all hi==0-guarded), so no
  // cross-lane exchange is needed at all.
  kout[0] = c3[0];
  kout[1] = c3[1];
  kout[2] = c3[2];
}

__global__ __launch_bounds__(TPB)
void ode_step_kernel(const float* __restrict__ W1, const float* __restrict__ b1,
                     const float* __restrict__ W2, const float* __restrict__ b2,
                     const float* __restrict__ W3, const float* __restrict__ b3,
                     float* __restrict__ ybuf0,   // d_out, stride 4
                     float* __restrict__ ybuf1,   // ws,    stride 3
                     float* __restrict__ partials,
                     const OdeState* __restrict__ st, int n)
{
  __shared__ float sW1[64*4], sB1[64], sW2[64*64], sB2[64], sW3[16*64], sB3[16];
  __shared__ float sX[NW][16*4];
  __shared__ float sH[NW][16*HSTRIDE];
  __shared__ float sRed[TPB];

  OdeState s = *st;
  if (s.done) return;                       // uniform: EXEC stays all-ones

  int tid = threadIdx.x;
  // Stage weights to LDS with vectorized copies (b128 global load + b128 ds store)
  for (int i = tid; i < 64;      i += TPB) ((float4*)sW1)[i] = ((const float4*)W1)[i];
  for (int i = tid; i < 64*16;   i += TPB) ((float4*)sW2)[i] = ((const float4*)W2)[i];
  for (int i = tid; i < 16;      i += TPB) {
    ((float4*)sB1)[i] = ((const float4*)b1)[i];
    ((float4*)sB2)[i] = ((const float4*)b2)[i];
  }
  for (int i = tid; i < 16*16;   i += TPB)
    ((float4*)sW3)[i] = (i < 48) ? ((const float4*)W3)[i]
                                 : make_float4(0.f, 0.f, 0.f, 0.f);
  for (int i = tid; i < 16;      i += TPB) sB3[i] = (i < 3) ? b3[i] : 0.0f;
  __syncthreads();

  int wave = tid >> 5, lane = tid & 31, ln = lane & 15, hi = lane >> 4;
  int e = blockIdx.x * EV_PER_BLOCK + wave*16 + ln;
  if (e >= n) e = n - 1;                    // clamp keeps EXEC full for WMMA

  // Hoist ALL A-fragments into registers: invariant across the 7 stages,
  // WMMA reads them in place (no copies). ~160 VGPRs.
  v2f aw1[4];
  v2f aw2[16][4];
  v2f aw3[16];
  #pragma unroll
  for (int m = 0; m < 4; ++m)
    aw1[m] = *(const v2f*)&sW1[(m*16 + ln)*4 + 2*hi];
  #pragma unroll
  for (int k = 0; k < 16; ++k) {
    #pragma unroll
    for (int m = 0; m < 4; ++m)
      aw2[k][m] = *(const v2f*)&sW2[(m*16 + ln)*64 + 4*k + 2*hi];
    aw3[k] = *(const v2f*)&sW3[ln*64 + 4*k + 2*hi];
  }

  const float* ysrc; int ss; float* ydst; int ds;
  if (s.cur == 0) { ysrc = ybuf0; ss = 4; ydst = ybuf1; ds = 3; }
  else            { ysrc = ybuf1; ss = 3; ydst = ybuf0; ds = 4; }

  float y0[3] = { ysrc[(size_t)e*ss+0], ysrc[(size_t)e*ss+1], ysrc[(size_t)e*ss+2] };
  float dt_e = fminf(s.dt, 1.0f - s.t);

  float k[7][3];
  #pragma unroll
  for (int i = 0; i < 7; ++i) {
    float yi[3] = { y0[0], y0[1], y0[2] };
    #pragma unroll
    for (int j = 0; j < 6; ++j) {
      if (j < i) {
        float a = d_A[i][j];
        yi[0] += dt_e * a * k[j][0];
        yi[1] += dt_e * a * k[j][1];
        yi[2] += dt_e * a * k[j][2];
      }
    }
    odefunc(s.t + d_C[i]*dt_e, yi, k[i], aw1, aw2, aw3,
            sB1, sB2, sB3, sX[wave], sH[wave], ln, hi);
  }

  float y5[3], err[3];
  #pragma unroll
  for (int c = 0; c < 3; ++c) {
    float s5 = 0.0f, se = 0.0f;
    #pragma unroll
    for (int i = 0; i < 7; ++i) { s5 += d_B5[i]*k[i][c]; se += d_E[i]*k[i][c]; }
    y5[c]  = y0[c] + dt_e * s5;
    err[c] = dt_e * se;
  }

  float local = 0.0f;
  #pragma unroll
  for (int c = 0; c < 3; ++c) {
    float sc = 1e-6f + 1e-5f * fmaxf(fabsf(y0[c]), fabsf(y5[c]));
    float q = err[c] / sc;
    local += q*q;
  }
  if (hi == 0) {                            // lanes 16-31 are duplicates
    ydst[(size_t)e*ds+0] = y5[0];
    ydst[(size_t)e*ds+1] = y5[1];
    ydst[(size_t)e*ds+2] = y5[2];
  } else {
    local = 0.0f;
  }

  sRed[tid] = local;
  __syncthreads();
  for (int off = TPB/2; off > 0; off >>= 1) {
    if (tid < off) sRed[tid] += sRed[tid+off];
    __syncthreads();
  }
  if (tid == 0) partials[blockIdx.x] = sRed[0];
}

__global__ __launch_bounds__(TPB)
void ode_reduce_kernel(const float* __restrict__ partials, OdeState* __restrict__ st,
                       int nblk, int n)
{
  __shared__ float sRed[TPB];
  OdeState s = *st;
  if (s.done) return;
  float acc = 0.0f;
  for (int i = threadIdx.x; i < nblk; i += TPB) acc += partials[i];  // fixed order
  sRed[threadIdx.x] = acc;
  __syncthreads();
  for (int off = TPB/2; off > 0; off >>= 1) {
    if (threadIdx.x < off) sRed[threadIdx.x] += sRed[threadIdx.x+off];
    __syncthreads();
  }
  if (threadIdx.x == 0) {
    float dt_e = fminf(s.dt, 1.0f - s.t);
    float en = sqrtf(sRed[0] / (float)(n*3));
    en = fmaxf(en, 1e-10f);
    bool accept = (en <= 1.0f);             // !done guaranteed here
    float t_n = accept ? (s.t + dt_e) : s.t;
    float fac = 0.9f * powf(en, -0.2f);
    fac = fminf(fmaxf(fac, 0.2f), 10.0f);
    OdeState ns;
    ns.t = t_n;
    ns.dt = dt_e * fac;
    ns.done = (t_n >= 1.0f - 1e-9f) ? 1u : 0u;
    ns.cur = accept ? (s.cur ^ 1u) : s.cur;
    *st = ns;
  }
}

__global__ void ode_init_kernel(const float* __restrict__ events, float* __restrict__ out,
                                OdeState* st, int n)
{
  int i = blockIdx.x*blockDim.x + threadIdx.x;
  if (i < n) {
    ((float4*)out)[i] = ((const float4*)events)[i];
  }
  if (i == 0) { st->t = 0.0f; st->dt = 0.05f; st->done = 0u; st->cur = 0u; }
}

__global__ void ode_final_kernel(const float* __restrict__ events,
                                 const float* __restrict__ ybuf1,
                                 float* __restrict__ out,
                                 const OdeState* __restrict__ st, int n)
{
  OdeState s = *st;
  int i = blockIdx.x*blockDim.x + threadIdx.x;
  if (i >= n) return;
  float y0, y1, y2;
  if (s.cur == 0) {
    y0 = out[(size_t)i*4+0]; y1 = out[(size_t)i*4+1]; y2 = out[(size_t)i*4+2];
  } else {
    y0 = ybuf1[(size_t)i*3+0]; y1 = ybuf1[(size_t)i*3+1]; y2 = ybuf1[(size_t)i*3+2];
  }
  float pol = events[(size_t)i*4+3];
  ((float4*)out)[i] = make_float4(y0, y1, y2, pol);
}

extern "C" void kernel_launch(void* const* d_in, const int* in_sizes, int n_in,
                              void* d_out, int out_size, void* d_ws, size_t ws_size,
                              hipStream_t stream)
{
  const float* events = (const float*)d_in[0];
  const float* W1 = (const float*)d_in[1];
  const float* b1 = (const float*)d_in[2];
  const float* W2 = (const float*)d_in[3];
  const float* b2 = (const float*)d_in[4];
  const float* W3 = (const float*)d_in[5];
  const float* b3 = (const float*)d_in[6];
  float* out = (float*)d_out;
  int n = in_sizes[0] / 4;
  int nblk = (n + EV_PER_BLOCK - 1) / EV_PER_BLOCK;

  // workspace layout: ybuf1 (n*3 f32) | partials (nblk f32) | OdeState (16B aligned)
  float* ybuf1 = (float*)d_ws;
  float* partials = ybuf1 + (size_t)n*3;
  OdeState* st = (OdeState*)(((uintptr_t)(partials + nblk) + 15u) & ~(uintptr_t)15u);

  int tpb = 256, g = (n + tpb - 1) / tpb;
  ode_init_kernel<<<g, tpb, 0, stream>>>(events, out, st, n);
  for (int step = 0; step < MAX_STEPS; ++step) {
    ode_step_kernel<<<nblk, TPB, 0, stream>>>(W1, b1, W2, b2, W3, b3,
                                              out, ybuf1, partials, st, n);
    ode_reduce_kernel<<<1, TPB, 0, stream>>>(partials, st, nblk, n);
  }
  ode_final_kernel<<<g, tpb, 0, stream>>>(events, ybuf1, out, st, n);
}